// EmbeddingReader_20091857011458
// MI455X (gfx1250) — compile-verified
//
#include <hip/hip_runtime.h>

typedef float v2f __attribute__((ext_vector_type(2)));
typedef float v8f __attribute__((ext_vector_type(8)));

#define M_TOT 8192
#define K_TOT 1024
#define N_TOT 32000
#define TILE  64           // block tile in M and N
#define KB    16           // K panel depth staged in LDS
#define PSTRIDE 160        // dwords per K-pair row: 128 data + 32 pad (bank split)

// Detect the gfx1250 async global->LDS copy builtins (ASYNCcnt-tracked).
#if defined(__has_builtin)
#if __has_builtin(__builtin_amdgcn_global_load_async_to_lds_b32) && \
    __has_builtin(__builtin_amdgcn_s_wait_asynccnt)
#define HAVE_ASYNC_LDS 1
#endif
#endif
#ifndef HAVE_ASYNC_LDS
#define HAVE_ASYNC_LDS 0
#endif

#if HAVE_ASYNC_LDS
typedef __attribute__((address_space(1))) int as1_int;   // global
typedef __attribute__((address_space(3))) int as3_int;   // LDS
#endif

// ---------------------------------------------------------------------------
// Kernel 1: embedding gather.  out[row, :] = in_weight[ids[row], :]
// 8192 blocks x 256 threads, one float4 (16B) per thread -> global_load_b128.
// ---------------------------------------------------------------------------
__global__ __launch_bounds__(256)
void embed_gather_kernel(const int* __restrict__ ids,
                         const float* __restrict__ W,   // [32000, 1024]
                         float* __restrict__ out)       // [8192, 1024]
{
    const int row = blockIdx.x;
    const int tok = ids[row];
    const float4* __restrict__ src = (const float4*)(W + (size_t)tok * K_TOT);
    float4* __restrict__ dst       = (float4*)(out + (size_t)row * K_TOT);
    dst[threadIdx.x] = src[threadIdx.x];
}

// ---------------------------------------------------------------------------
// Kernel 2: logits GEMM  C[8192,32000] = A[8192,1024] * B[1024,32000], fp32.
// 128 threads = 4 wave32; block computes a 64x64 C tile, wave w owns a 16-row
// strip.  B panel staged in LDS K-pair interleaved: element (k, n) at
//   ldsB[(k>>1)*PSTRIDE + 2n + (k&1)]
// so a B fragment for V_WMMA_F32_16X16X4_F32 is one aligned ds_load_b64
// (compiler fuses pairs into ds_load_2addr_b64).  PSTRIDE=160 (mod 64 = 32
// banks) puts the two lane halves on disjoint bank halves -> conflict-free.
// B panel is fetched with global_load_async_to_lds_b32 (ASYNCcnt) when the
// toolchain exposes it; global addresses are incremented, not recomputed.
// ---------------------------------------------------------------------------
__global__ __launch_bounds__(128)
void unembed_gemm_kernel(const float* __restrict__ A,   // [8192, 1024]
                         const float* __restrict__ Bw,  // [1024, 32000]
                         float* __restrict__ C)         // [8192, 32000]
{
    __shared__ float ldsB[(KB / 2) * PSTRIDE];          // 5120 B

    const int tid  = threadIdx.x;     // 0..127
    const int lane = tid & 31;
    const int wave = tid >> 5;        // 0..3
    const int hf   = lane >> 4;       // lane half: 0 or 1
    const int l16  = lane & 15;

    const int blockM = blockIdx.y * TILE;
    const int blockN = blockIdx.x * TILE;

    // A fragment: lane L holds row M = L % 16; half 0 -> K={kk,kk+1}, half 1 ->
    // K={kk+2,kk+3}  ==> one global_load_b64 per lane per kk step.
    const float* __restrict__ Arow = A + (size_t)(blockM + wave * 16 + l16) * K_TOT;

    v8f acc[4];
    #pragma unroll
    for (int nt = 0; nt < 4; ++nt) acc[nt] = (v8f){0.f,0.f,0.f,0.f,0.f,0.f,0.f,0.f};

    // Loader mapping: panel = KB*64 = 1024 dwords, 8 per thread.
    // idx = j*128 + tid ; k = idx/64 ; n = idx%64  (consecutive tids ->
    // consecutive n: coalesced 256B rows on the global side).
    // Precompute global/LDS addresses once; bump global ptrs per panel.
    const float* gptr[8];
    float*       lptr[8];
    #pragma unroll
    for (int j = 0; j < 8; ++j) {
        const int idx = j * 128 + tid;
        const int k = idx >> 6;
        const int n = idx & 63;
        gptr[j] = Bw + (size_t)k * N_TOT + blockN + n;
        lptr[j] = &ldsB[(k >> 1) * PSTRIDE + 2 * n + (k & 1)];
    }

    for (int k0 = 0; k0 < K_TOT; k0 += KB) {
        __syncthreads();   // previous panel fully consumed

#if HAVE_ASYNC_LDS
        #pragma unroll
        for (int j = 0; j < 8; ++j) {
            __builtin_amdgcn_global_load_async_to_lds_b32(
                (as1_int*)gptr[j], (as3_int*)lptr[j], 0, 0);
            gptr[j] += (size_t)KB * N_TOT;   // advance to next panel
        }
        if (k0 + KB < K_TOT) {
            __builtin_prefetch(gptr[0], 0, 1);   // warm next panel in L2
        }
        __builtin_amdgcn_s_wait_asynccnt(0);
#else
        // Fallback: register staging into the same interleaved layout.
        #pragma unroll
        for (int j = 0; j < 8; ++j) {
            *lptr[j] = *gptr[j];
            gptr[j] += (size_t)KB * N_TOT;
        }
        if (k0 + KB < K_TOT) {
            __builtin_prefetch(gptr[0], 0, 1);
        }
#endif
        __syncthreads();   // panel visible to all waves

        #pragma unroll
        for (int kk = 0; kk < KB; kk += 4) {
            const v2f afrag = *(const v2f*)(Arow + k0 + kk + hf * 2);
            const float* brow = &ldsB[(kk / 2 + hf) * PSTRIDE + 2 * l16];
            #pragma unroll
            for (int nt = 0; nt < 4; ++nt) {
                const v2f bfrag = *(const v2f*)(brow + nt * 32);  // ds_load_b64
                acc[nt] = __builtin_amdgcn_wmma_f32_16x16x4_f32(
                    false, afrag, false, bfrag, (short)0, acc[nt], false, false);
            }
        }
    }

    // Epilogue: C/D layout -> M = r + 8*hf, N = lane%16 for accumulator VGPR r.
    const int cm = blockM + wave * 16 + hf * 8;
    const int cn = blockN + l16;
    #pragma unroll
    for (int nt = 0; nt < 4; ++nt) {
        float* __restrict__ cp = C + (size_t)cm * N_TOT + cn + nt * 16;
        #pragma unroll
        for (int r = 0; r < 8; ++r) {
            cp[(size_t)r * N_TOT] = acc[nt][r];
        }
    }
}

// ---------------------------------------------------------------------------
// Launch.  Inputs (setup_inputs order): [0] inputs i32[4,2048], [1] embeds
// f32[4,2048,1024], [2] in_weight f32[32000,1024], [3] out_weight f32[1024,32000].
// d_out = embedded (8192*1024 f32) || logits (8192*32000 f32).
// ---------------------------------------------------------------------------
extern "C" void kernel_launch(void* const* d_in, const int* in_sizes, int n_in,
                              void* d_out, int out_size, void* d_ws, size_t ws_size,
                              hipStream_t stream) {
    const int*   ids     = (const int*)d_in[0];
    const float* embeds  = (const float*)d_in[1];
    const float* in_w    = (const float*)d_in[2];
    const float* out_w   = (const float*)d_in[3];

    float* embedded = (float*)d_out;                          // [8192, 1024]
    float* logits   = (float*)d_out + (size_t)M_TOT * K_TOT;  // [8192, 32000]

    embed_gather_kernel<<<M_TOT, 256, 0, stream>>>(ids, in_w, embedded);

    dim3 grid(N_TOT / TILE, M_TOT / TILE);  // (500, 128)
    unembed_gemm_kernel<<<grid, 128, 0, stream>>>(embeds, out_w, logits);
}